// BaselineGNN_11974368821436
// MI455X (gfx1250) — compile-verified
//
#include <hip/hip_runtime.h>

#define N_NODES   100000
#define N_EDGES   1600000
#define N_GRAPHS  1024
#define IN_LOCAL  16
#define IN_GLOBAL 8
#define HIDDEN    128
#define KC        (IN_LOCAL + HIDDEN)   // 144

typedef __attribute__((ext_vector_type(2))) float v2f;
typedef __attribute__((ext_vector_type(8))) float v8f;

#define LDSA_STRIDE 145   // odd stride -> conflict-free column reads
#define LDSH_STRIDE 129

// ---------------------------------------------------------------------------
// Kernel 1: hg[g,h] = relu(global_feat[g,:] @ Wg[:,h] + bg[h])   [1024 x 128]
// ---------------------------------------------------------------------------
__global__ __launch_bounds__(256) void k_global_enc(
    const float* __restrict__ gf, const float* __restrict__ Wg,
    const float* __restrict__ bg, float* __restrict__ hg) {
  int idx = blockIdx.x * 256 + threadIdx.x;     // 131072 threads exactly
  int g = idx >> 7, h = idx & 127;
  float acc = bg[h];
#pragma unroll
  for (int k = 0; k < IN_GLOBAL; ++k)
    acc = fmaf(gf[g * IN_GLOBAL + k], Wg[k * HIDDEN + h], acc);
  hg[idx] = fmaxf(acc, 0.0f);
}

// ---------------------------------------------------------------------------
// Kernel 2: fused node GEMMs, one 16-node tile per 256-thread block (8 waves).
//   h0  = relu([x | hg[batch]] @ Wc + bc)      (K=144, via LDS-staged A)
//   m   = relu(h0 @ Wm + bm)                   -> m_buf
//   acc = m + (h0 @ Ws + bs)                   -> acc_buf (self-loop + Ws path)
// Each wave owns one 16x16 output column tile; v_wmma_f32_16x16x4_f32 chains.
// ---------------------------------------------------------------------------
__global__ __launch_bounds__(256) void k_node_gemms(
    const float* __restrict__ x,  const int* __restrict__ batch,
    const float* __restrict__ hg,
    const float* __restrict__ Wc, const float* __restrict__ bc,
    const float* __restrict__ Wm, const float* __restrict__ bm,
    const float* __restrict__ Ws, const float* __restrict__ bs,
    float* __restrict__ m_buf, float* __restrict__ acc_buf) {
  __shared__ float lds_a[16 * LDSA_STRIDE];   // [x | hg[batch]] tile, 16 x 144
  __shared__ float lds_h[16 * LDSH_STRIDE];   // h0 tile, 16 x 128

  const int tid = threadIdx.x;
  const int nodeBase = blockIdx.x * 16;       // 6250 blocks * 16 = 100000 exact

  // ---- stage A = [x | hg[batch]] into LDS (coalesced-ish cooperative fill)
  for (int i = tid; i < 16 * KC; i += 256) {
    int row = i / KC, k = i - row * KC;
    int node = nodeBase + row;
    float v = (k < IN_LOCAL) ? x[node * IN_LOCAL + k]
                             : hg[batch[node] * HIDDEN + (k - IN_LOCAL)];
    lds_a[row * LDSA_STRIDE + k] = v;
  }
  __syncthreads();

  const int lane  = tid & 31;
  const int wave  = tid >> 5;
  const int half  = lane >> 4;     // selects K pair {0,1} vs {2,3} in fragments
  const int idx16 = lane & 15;     // A row / B,C,D column within tile
  const int colBase = wave * 16;   // 8 waves cover 128 output columns

  // ---- h0 tile: K = 144, 36 WMMA f32 16x16x4 steps
  v8f acc = {};
#pragma unroll 4
  for (int kb = 0; kb < KC; kb += 4) {
    const int k0 = kb + 2 * half;
    v2f a, b;
    a.x = lds_a[idx16 * LDSA_STRIDE + k0];
    a.y = lds_a[idx16 * LDSA_STRIDE + k0 + 1];
    b.x = Wc[k0 * HIDDEN + colBase + idx16];
    b.y = Wc[(k0 + 1) * HIDDEN + colBase + idx16];
    acc = __builtin_amdgcn_wmma_f32_16x16x4_f32(
        false, a, false, b, (short)0, acc, false, false);
  }
  {
    const float bias = bc[colBase + idx16];
#pragma unroll
    for (int j = 0; j < 8; ++j) {
      int row = j + 8 * half;      // C/D layout: VGPR j -> M = j (+8 for hi lanes)
      lds_h[row * LDSH_STRIDE + colBase + idx16] = fmaxf(acc[j] + bias, 0.0f);
    }
  }
  __syncthreads();

  // ---- m and s tiles: K = 128, shared A fragments, two WMMA chains
  v8f am = {}, as = {};
#pragma unroll 4
  for (int kb = 0; kb < HIDDEN; kb += 4) {
    const int k0 = kb + 2 * half;
    v2f a, b0, b1;
    a.x  = lds_h[idx16 * LDSH_STRIDE + k0];
    a.y  = lds_h[idx16 * LDSH_STRIDE + k0 + 1];
    b0.x = Wm[k0 * HIDDEN + colBase + idx16];
    b0.y = Wm[(k0 + 1) * HIDDEN + colBase + idx16];
    b1.x = Ws[k0 * HIDDEN + colBase + idx16];
    b1.y = Ws[(k0 + 1) * HIDDEN + colBase + idx16];
    am = __builtin_amdgcn_wmma_f32_16x16x4_f32(
        false, a, false, b0, (short)0, am, false, false);
    as = __builtin_amdgcn_wmma_f32_16x16x4_f32(
        false, a, false, b1, (short)0, as, false, false);
  }
  {
    const int col = colBase + idx16;
    const float biasm = bm[col], biass = bs[col];
#pragma unroll
    for (int j = 0; j < 8; ++j) {
      int row  = j + 8 * half;
      int node = nodeBase + row;
      float mv = fmaxf(am[j] + biasm, 0.0f);
      m_buf[(size_t)node * HIDDEN + col]   = mv;
      acc_buf[(size_t)node * HIDDEN + col] = mv + as[j] + biass;  // m + s + bs
    }
  }
}

// ---------------------------------------------------------------------------
// Kernel 3: edge scatter  acc[dst] += m[src]  (wave per edge, 4 floats/lane)
// This is the HBM/atomic-bound pass (~1.6-2 GB of traffic).
// ---------------------------------------------------------------------------
__global__ __launch_bounds__(256) void k_scatter(
    const int* __restrict__ edge_index, const float* __restrict__ m_buf,
    float* __restrict__ acc_buf) {
  const int lane = threadIdx.x & 31;
  const int e = blockIdx.x * 8 + (threadIdx.x >> 5);   // 200000 blocks * 8 exact
  const int src = edge_index[e];
  const int dst = edge_index[N_EDGES + e];
  const float* ms = m_buf   + (size_t)src * HIDDEN;
  float*       ad = acc_buf + (size_t)dst * HIDDEN;
#pragma unroll
  for (int t = 0; t < 4; ++t) {
    int c = lane + t * 32;
    unsafeAtomicAdd(&ad[c], ms[c]);   // -> global_atomic_add_f32
  }
}

// ---------------------------------------------------------------------------
// Kernel 4: out[node,:] = relu(acc[node,:]) @ Wo + bo   (wave per node)
// ---------------------------------------------------------------------------
__global__ __launch_bounds__(256) void k_out(
    const float* __restrict__ acc_buf, const float* __restrict__ Wo,
    const float* __restrict__ bo, float* __restrict__ out) {
  const int lane = threadIdx.x & 31;
  const int node = blockIdx.x * 8 + (threadIdx.x >> 5);  // 12500 blocks * 8 exact
  float a0 = 0.0f, a1 = 0.0f;
#pragma unroll
  for (int t = 0; t < 4; ++t) {
    int k = lane + t * 32;
    float h = fmaxf(acc_buf[(size_t)node * HIDDEN + k], 0.0f);
    a0 = fmaf(h, Wo[k * 2 + 0], a0);
    a1 = fmaf(h, Wo[k * 2 + 1], a1);
  }
#pragma unroll
  for (int off = 16; off >= 1; off >>= 1) {
    a0 += __shfl_xor(a0, off, 32);
    a1 += __shfl_xor(a1, off, 32);
  }
  if (lane == 0) {
    out[node * 2 + 0] = a0 + bo[0];
    out[node * 2 + 1] = a1 + bo[1];
  }
}

// ---------------------------------------------------------------------------
extern "C" void kernel_launch(void* const* d_in, const int* in_sizes, int n_in,
                              void* d_out, int out_size, void* d_ws, size_t ws_size,
                              hipStream_t stream) {
  (void)in_sizes; (void)n_in; (void)out_size; (void)ws_size;
  const float* x     = (const float*)d_in[0];
  const int*   edge  = (const int*)  d_in[1];   // [2, N_EDGES]: row0=src, row1=dst
  const int*   batch = (const int*)  d_in[2];
  const float* gf    = (const float*)d_in[3];
  const float* Wg    = (const float*)d_in[4];
  const float* bg    = (const float*)d_in[5];
  const float* Wc    = (const float*)d_in[6];
  const float* bc    = (const float*)d_in[7];
  const float* Wm    = (const float*)d_in[8];
  const float* bm    = (const float*)d_in[9];
  const float* Ws    = (const float*)d_in[10];
  const float* bs    = (const float*)d_in[11];
  const float* Wo    = (const float*)d_in[12];
  const float* bo    = (const float*)d_in[13];
  float* out = (float*)d_out;

  // workspace layout (floats): hg | m_buf | acc_buf  (~103 MB total)
  float* hg      = (float*)d_ws;
  float* m_buf   = hg    + (size_t)N_GRAPHS * HIDDEN;
  float* acc_buf = m_buf + (size_t)N_NODES  * HIDDEN;

  k_global_enc<<<(N_GRAPHS * HIDDEN) / 256, 256, 0, stream>>>(gf, Wg, bg, hg);
  k_node_gemms<<<N_NODES / 16, 256, 0, stream>>>(x, batch, hg, Wc, bc,
                                                 Wm, bm, Ws, bs, m_buf, acc_buf);
  k_scatter<<<N_EDGES / 8, 256, 0, stream>>>(edge, m_buf, acc_buf);
  k_out<<<N_NODES / 8, 256, 0, stream>>>(acc_buf, Wo, bo, out);
}